// HyperGNN_10849087390587
// MI455X (gfx1250) — compile-verified
//
#include <hip/hip_runtime.h>

// HGCN on gfx1250 (MI455X):
//  - f32 WMMA (V_WMMA_F32_16X16X4_F32) for the mobius matvecs, fully fused
//    logmap0 -> GEMM -> expmap0 -> proj -> mobius_add -> proj epilogue.
//  - Edge aggregation done dst-centric: a CSR of the edge list is built once
//    per launch (800k int atomics + one block-scan) and each layer's mean
//    aggregation is a zero-atomic, register-accumulating gather fused with
//    proj(expmap0(.)) and the leaky-relu hyperbolic activation.
//  - Node features (12.8MB) fit in the 192MB L2, so gathers stay on-chip.

#define NN 50000
#define NE 800000
#define NG 512
#define DD 64
#define EPSF 1e-7f
#define MAXN (1.0f - 1e-5f)

typedef __attribute__((ext_vector_type(2))) float v2f;
typedef __attribute__((ext_vector_type(8))) float v8f;

__device__ __forceinline__ float wsum(float v) {
#pragma unroll
  for (int o = 16; o > 0; o >>= 1) v += __shfl_xor(v, o, 32);
  return v;
}

// ---------------- CSR build (once per launch) ----------------

__global__ void k_hist(const int* __restrict__ ei, int* __restrict__ deg) {
  int e = blockIdx.x * blockDim.x + threadIdx.x;
  if (e < NE) atomicAdd(deg + ei[NE + e], 1);
}

__global__ __launch_bounds__(1024) void k_scan(const int* __restrict__ deg,
                                               int* __restrict__ rowptr,
                                               int* __restrict__ cursor) {
  __shared__ int sm[1024];
  const int t = threadIdx.x;
  const int CH = (NN + 1023) / 1024;
  int lo = t * CH;
  int hi = lo + CH;
  if (lo > NN) lo = NN;
  if (hi > NN) hi = NN;
  int local = 0;
  for (int i = lo; i < hi; ++i) local += deg[i];
  sm[t] = local;
  __syncthreads();
  for (int d = 1; d < 1024; d <<= 1) {
    int add = (t >= d) ? sm[t - d] : 0;
    __syncthreads();
    sm[t] += add;
    __syncthreads();
  }
  int off = sm[t] - local;  // exclusive prefix of this chunk
  for (int i = lo; i < hi; ++i) {
    rowptr[i] = off;
    cursor[i] = off;
    off += deg[i];
  }
  if (t == 1023) rowptr[NN] = sm[1023];
}

__global__ void k_fill(const int* __restrict__ ei, int* __restrict__ cursor,
                       int* __restrict__ srclist) {
  int e = blockIdx.x * blockDim.x + threadIdx.x;
  if (e >= NE) return;
  int d = ei[NE + e];
  int pos = atomicAdd(cursor + d, 1);
  srclist[pos] = ei[e];
}

// ---------------- elementwise Poincare maps ----------------

// h = proj(expmap0(x)) ; one wave per node, 2 elems per lane
__global__ void k_expmap_proj(const float* __restrict__ x, float* __restrict__ h, int n) {
  int w = blockIdx.x * (blockDim.x >> 5) + (threadIdx.x >> 5);
  int lane = threadIdx.x & 31;
  if (w >= n) return;
  float2 v = ((const float2*)(x + (size_t)w * DD))[lane];
  float nrm = sqrtf(wsum(v.x * v.x + v.y * v.y));
  float nl = fmaxf(nrm, EPSF);
  float s = tanhf(nl) / nl;
  float ne = s * nrm;
  if (ne > MAXN) s *= MAXN / ne;
  float2 o;
  o.x = s * v.x; o.y = s * v.y;
  ((float2*)(h + (size_t)w * DD))[lane] = o;
}

// t = logmap0(h)
__global__ void k_logmap(const float* __restrict__ h, float* __restrict__ t, int n) {
  int w = blockIdx.x * (blockDim.x >> 5) + (threadIdx.x >> 5);
  int lane = threadIdx.x & 31;
  if (w >= n) return;
  float2 v = ((const float2*)(h + (size_t)w * DD))[lane];
  float nrm = sqrtf(wsum(v.x * v.x + v.y * v.y));
  float nl = fmaxf(nrm, EPSF);
  float s = atanhf(fminf(nl, MAXN)) / nl;
  float2 o;
  o.x = s * v.x; o.y = s * v.y;
  ((float2*)(t + (size_t)w * DD))[lane] = o;
}

// ---------------- fused mean-aggregation (no atomics) ----------------
// h[w] = proj(expmap0( mean_{s in N(w)} T[s] ))  [+ fused hyp_act leaky 0.2]
template <int ACT>
__global__ void k_agg_gather(const float* __restrict__ T, const int* __restrict__ rowptr,
                             const int* __restrict__ srclist, float* __restrict__ h) {
  int w = blockIdx.x * (blockDim.x >> 5) + (threadIdx.x >> 5);
  int lane = threadIdx.x & 31;
  if (w >= NN) return;
  int beg = rowptr[w], end = rowptr[w + 1];
  float ax = 0.f, ay = 0.f;
  for (int j0 = beg; j0 < end; j0 += 32) {
    int sidx = (j0 + lane < end) ? srclist[j0 + lane] : 0;
    int nj = end - j0;
    if (nj > 32) nj = 32;
    for (int u = 0; u < nj; ++u) {
      int s = __shfl(sidx, u, 32);
      float2 v = ((const float2*)(T + (size_t)s * DD))[lane];
      ax += v.x;
      ay += v.y;
    }
  }
  float c = fmaxf((float)(end - beg), 1.0f);
  float vx = ax / c, vy = ay / c;
  float nrm = sqrtf(wsum(vx * vx + vy * vy));
  float nl = fmaxf(nrm, EPSF);
  float s = tanhf(nl) / nl;
  float ne = s * nrm;
  if (ne > MAXN) s *= MAXN / ne;  // h1 = s*v
  float2 o;
  if (ACT) {
    float n2 = fmaxf(s * nrm, EPSF);  // |h1|
    float k = s * (atanhf(fminf(n2, MAXN)) / n2);
    float tx = k * vx, ty = k * vy;
    tx = tx >= 0.0f ? tx : 0.2f * tx;
    ty = ty >= 0.0f ? ty : 0.2f * ty;
    float n3 = sqrtf(wsum(tx * tx + ty * ty));
    float nl3 = fmaxf(n3, EPSF);
    float s3 = tanhf(nl3) / nl3;
    float ne3 = s3 * n3;
    if (ne3 > MAXN) s3 *= MAXN / ne3;
    o.x = s3 * tx; o.y = s3 * ty;
  } else {
    o.x = s * vx; o.y = s * vy;
  }
  ((float2*)(h + (size_t)w * DD))[lane] = o;
}

// ---------------- pooling ----------------

__global__ void k_pool_scatter(const float* __restrict__ h, const int* __restrict__ batch,
                               float* __restrict__ pooled, float* __restrict__ pcnt) {
  int w = blockIdx.x * (blockDim.x >> 5) + (threadIdx.x >> 5);
  int lane = threadIdx.x & 31;
  if (w >= NN) return;
  float2 v = ((const float2*)(h + (size_t)w * DD))[lane];
  float nrm = sqrtf(wsum(v.x * v.x + v.y * v.y));
  float nl = fmaxf(nrm, EPSF);
  float s = atanhf(fminf(nl, MAXN)) / nl;
  int g = batch[w];
  float* pr = pooled + (size_t)g * DD + lane * 2;
  atomicAdd(pr, s * v.x);
  atomicAdd(pr + 1, s * v.y);
  if (lane == 0) atomicAdd(pcnt + g, 1.0f);
}

__global__ void k_pool_finalize(float* __restrict__ pooled, const float* __restrict__ pcnt, int n) {
  int w = blockIdx.x * (blockDim.x >> 5) + (threadIdx.x >> 5);
  int lane = threadIdx.x & 31;
  if (w >= n) return;
  float c = fmaxf(pcnt[w], 1.0f);
  float2 v = ((float2*)(pooled + (size_t)w * DD))[lane];
  v.x /= c; v.y /= c;
  float nrm = sqrtf(wsum(v.x * v.x + v.y * v.y));
  float nl = fmaxf(nrm, EPSF);
  float s = tanhf(nl) / nl;
  float ne = s * nrm;
  if (ne > MAXN) s *= MAXN / ne;
  float2 o;
  o.x = s * v.x; o.y = s * v.y;
  ((float2*)(pooled + (size_t)w * DD))[lane] = o;
}

// ---------------- fused hyp_linear with f32 WMMA ----------------
// out = proj(mobius_add(proj(expmap0(logmap0(h) @ W^T)), proj(expmap0(b))))
// 128 threads (4 waves), 64 rows per block; each wave: 16 rows x OUT cols,
// K=64 in 16 steps of V_WMMA_F32_16X16X4_F32.
template <int OUT>
__global__ __launch_bounds__(128) void k_hyp_linear(const float* __restrict__ h,
                                                    const float* __restrict__ W,
                                                    const float* __restrict__ b,
                                                    float* __restrict__ out, int nrows) {
  __shared__ float tS[64][65];       // logmap0(h) tile
  __shared__ float wS[OUT][65];      // W (row o, col k)
  __shared__ float uS[64][OUT + 1];  // matmul result tile
  __shared__ float hbS[OUT];
  __shared__ float C1[64], C2[64];
  __shared__ float y2S;

  const int tid = threadIdx.x;
  const int rb = blockIdx.x * 64;

  for (int i = tid; i < 64 * 64 / 4; i += 128) {
    int row = (i * 4) / 64, col = (i * 4) % 64;
    int rg = rb + row;
    float4 v = make_float4(0.f, 0.f, 0.f, 0.f);
    if (rg < nrows) v = ((const float4*)(h + (size_t)rg * 64))[col >> 2];
    tS[row][col] = v.x; tS[row][col + 1] = v.y; tS[row][col + 2] = v.z; tS[row][col + 3] = v.w;
  }
  for (int i = tid; i < OUT * 64 / 4; i += 128) {
    int row = (i * 4) / 64, col = (i * 4) % 64;
    float4 v = ((const float4*)(W + (size_t)row * 64))[col >> 2];
    wS[row][col] = v.x; wS[row][col + 1] = v.y; wS[row][col + 2] = v.z; wS[row][col + 3] = v.w;
  }
  if (tid == 0) {
    float ss = 0.f;
    for (int c = 0; c < OUT; ++c) { float bv = b[c]; ss += bv * bv; }
    float nrm = sqrtf(ss), nl = fmaxf(nrm, EPSF);
    float s = tanhf(nl) / nl;
    float ne = s * nrm;
    if (ne > MAXN) s *= MAXN / ne;
    float y2 = 0.f;
    for (int c = 0; c < OUT; ++c) { float hv = s * b[c]; hbS[c] = hv; y2 += hv * hv; }
    y2S = y2;
  }
  __syncthreads();

  if (tid < 64) {  // in-place logmap0 scaling of rows
    float ss = 0.f;
    for (int c = 0; c < 64; ++c) { float v = tS[tid][c]; ss += v * v; }
    float nrm = sqrtf(ss), nl = fmaxf(nrm, EPSF);
    float s = atanhf(fminf(nl, MAXN)) / nl;
    for (int c = 0; c < 64; ++c) tS[tid][c] *= s;
  }
  __syncthreads();

  constexpr int NGROUP = OUT / 16;
  const int wv = tid >> 5;
  const int lane = tid & 31;
  const int lrow = lane & 15;
  const int half = lane >> 4;
  const int koff = half * 2;

  v8f accv[NGROUP];
  v8f zv = {0.f, 0.f, 0.f, 0.f, 0.f, 0.f, 0.f, 0.f};
#pragma unroll
  for (int g = 0; g < NGROUP; ++g) accv[g] = zv;

  for (int k = 0; k < 64; k += 4) {
    v2f A;
    A.x = tS[16 * wv + lrow][k + koff];
    A.y = tS[16 * wv + lrow][k + koff + 1];
#pragma unroll
    for (int g = 0; g < NGROUP; ++g) {
      v2f B;
      B.x = wS[16 * g + lrow][k + koff];  // B[k'][n] = W[n][k']
      B.y = wS[16 * g + lrow][k + koff + 1];
      accv[g] = __builtin_amdgcn_wmma_f32_16x16x4_f32(
          false, A, false, B, (short)0, accv[g], false, false);
    }
  }

#pragma unroll
  for (int g = 0; g < NGROUP; ++g)
#pragma unroll
    for (int r = 0; r < 8; ++r) {
      int node = 16 * wv + r + 8 * half;
      uS[node][16 * g + lrow] = accv[g][r];
    }
  __syncthreads();

  if (tid < 64) {  // per-row scalar epilogue
    float ssu = 0.f, dot = 0.f;
    for (int c = 0; c < OUT; ++c) { float u = uS[tid][c]; ssu += u * u; dot += u * hbS[c]; }
    float nu = sqrtf(ssu), nl = fmaxf(nu, EPSF);
    float te = tanhf(nl) / nl;
    float ne = te * nu;
    float s1 = (ne > MAXN) ? (MAXN / ne) : 1.0f;
    float cx = te * s1;
    float x2 = cx * cx * ssu;
    float xy = cx * dot;
    float y2 = y2S;
    float den = fmaxf(1.0f + 2.0f * xy + x2 * y2, EPSF);
    float c1 = (1.0f + 2.0f * xy + y2) * cx / den;
    float c2 = (1.0f - x2) / den;  // out = c1*u + c2*hb
    float nr2 = c1 * c1 * ssu + 2.0f * c1 * c2 * dot + c2 * c2 * y2;
    float nr = fmaxf(sqrtf(fmaxf(nr2, 0.0f)), EPSF);
    float c3 = (nr > MAXN) ? (MAXN / nr) : 1.0f;
    C1[tid] = c1 * c3;
    C2[tid] = c2 * c3;
  }
  __syncthreads();

  for (int i = tid; i < 64 * OUT / 4; i += 128) {
    int row = (i * 4) / OUT, col = (i * 4) % OUT;
    int rg = rb + row;
    if (rg < nrows) {
      float a = C1[row], bb = C2[row];
      float4 o;
      o.x = a * uS[row][col] + bb * hbS[col];
      o.y = a * uS[row][col + 1] + bb * hbS[col + 1];
      o.z = a * uS[row][col + 2] + bb * hbS[col + 2];
      o.w = a * uS[row][col + 3] + bb * hbS[col + 3];
      ((float4*)(out + (size_t)rg * OUT))[col >> 2] = o;
    }
  }
}

// ---------------- launch ----------------

extern "C" void kernel_launch(void* const* d_in, const int* in_sizes, int n_in,
                              void* d_out, int out_size, void* d_ws, size_t ws_size,
                              hipStream_t stream) {
  const float* x  = (const float*)d_in[0];
  const float* W1 = (const float*)d_in[1];
  const float* b1 = (const float*)d_in[2];
  const float* W2 = (const float*)d_in[3];
  const float* b2 = (const float*)d_in[4];
  const float* W3 = (const float*)d_in[5];
  const float* b3 = (const float*)d_in[6];
  const float* W4 = (const float*)d_in[7];
  const float* b4 = (const float*)d_in[8];
  const int* ei    = (const int*)d_in[9];
  const int* batch = (const int*)d_in[10];
  float* out = (float*)d_out;

  float* ws = (float*)d_ws;
  const size_t N64 = (size_t)NN * DD;
  float* A = ws;                        // current node features / tangent vectors
  float* B = ws + N64;                  // hyp_linear output
  float* pooled = ws + 2 * N64;         // [NG, 64]
  float* pcnt = pooled + (size_t)NG * DD;
  int* deg     = (int*)(pcnt + NG);
  int* rowptr  = deg + NN;              // NN+1
  int* cursor  = rowptr + NN + 1;
  int* srclist = cursor + NN;           // NE

  const int gw = (NN + 7) / 8;          // wave-per-node, 8 waves/block
  const int geT = (NE + 255) / 256;     // thread-per-edge
  const int gl = (NN + 63) / 64;        // 64 rows per hyp_linear block

  // CSR(by-dst) build, reused by all 3 layers
  hipMemsetAsync(deg, 0, NN * sizeof(int), stream);
  k_hist<<<geT, 256, 0, stream>>>(ei, deg);
  k_scan<<<1, 1024, 0, stream>>>(deg, rowptr, cursor);
  k_fill<<<geT, 256, 0, stream>>>(ei, cursor, srclist);

  k_expmap_proj<<<gw, 256, 0, stream>>>(x, A, NN);

  const float* Ws[3] = {W1, W2, W3};
  const float* bs[3] = {b1, b2, b3};
  for (int l = 0; l < 3; ++l) {
    k_hyp_linear<64><<<gl, 128, 0, stream>>>(A, Ws[l], bs[l], B, NN);
    k_logmap<<<gw, 256, 0, stream>>>(B, A, NN);
    if (l < 2) k_agg_gather<1><<<gw, 256, 0, stream>>>(A, rowptr, srclist, B);
    else       k_agg_gather<0><<<gw, 256, 0, stream>>>(A, rowptr, srclist, B);
    // result of aggregation is in B; swap roles for next layer
    float* tmp = A; A = B; B = tmp;
  }

  hipMemsetAsync(pooled, 0, ((size_t)NG * DD + NG) * sizeof(float), stream);
  k_pool_scatter<<<gw, 256, 0, stream>>>(A, batch, pooled, pcnt);
  k_pool_finalize<<<(NG + 7) / 8, 256, 0, stream>>>(pooled, pcnt, NG);
  k_hyp_linear<16><<<(NG + 63) / 64, 128, 0, stream>>>(pooled, W4, b4, out, NG);
}